// CWTConvNet_15685220565166
// MI455X (gfx1250) — compile-verified
//
#include <hip/hip_runtime.h>
#include <hip/hip_bf16.h>

// CWT conv bank on MI455X (gfx1250, wave32, WMMA, warp-specialized LDS staging).
// Per (b,c) block: Y[112 x 72] = W[112 x K] * Xpatch[K x 72],
// Xpatch[k,t] = x[t - pad + k] (implicit im2col from a small LDS line buffer).
// FP32 matrix path: v_wmma_f32_16x16x4_f32 (exact-precision, and this kernel is
// latency/bandwidth bound at ~3.5 GFLOP, so lower precision buys nothing).
//
// Waves 0-4: compute (one 16-wide N-tile each, loop over 7 M-tiles).
// Waves 5-7: double-buffer the next M-tile's 16xK W chunk global->LDS.

typedef __attribute__((ext_vector_type(2))) float v2f;
typedef __attribute__((ext_vector_type(8))) float v8f;

#define LCONV   2048
#define NFILT   112
#define TOUT    224
#define NT      80            // 72 valid cols padded to 5 tiles of 16
#define XBUF_N  768           // >= K + NT (K=561 -> 641), zero-padded tail
#define KMAX    568           // static bound for W chunk rows (K=561)
#define MTILES  7             // 112 / 16
#define NTHR    256           // 8 waves
#define NCOMP   5             // compute waves (= N-tiles)

template <int KC>
__global__ __launch_bounds__(NTHR) void cwt_wmma_kernel(
    const float* __restrict__ x,     // [BC, 2048]
    const float* __restrict__ w,     // [112, K] row-major
    float* __restrict__ out,         // [BC, 112, 224]
    int Kdyn, int pad)
{
    const int K = KC ? KC : Kdyn;    // compile-time for the hot instantiation

    __shared__ float xbuf[XBUF_N];                       // zero-padded input line
    __shared__ __align__(16) float wbuf[2][16 * KMAX];   // W chunk double buffer
    __shared__ float ytile[NFILT][NT];                   // f32 result tile

    const int bc  = blockIdx.x;
    const int tid = threadIdx.x;
    const float* xg = x + (size_t)bc * LCONV;

    // ---- phase 0: stage input line + W chunk 0 --------------------------
    for (int i = tid; i < XBUF_N; i += NTHR) {
        const int g = i - pad;
        float v = 0.0f;
        if (i < K + NT && g >= 0 && g < LCONV) v = xg[g];
        xbuf[i] = v;
    }
    {   // chunk 0 = W rows 0..15, contiguous 16*K floats (16K % 4 == 0)
        const int n4 = 4 * K;                            // float4 count
        const float4* src = (const float4*)w;
        float4* dst = (float4*)&wbuf[0][0];
        for (int i = tid; i < n4; i += NTHR) dst[i] = src[i];
    }
    __syncthreads();

    // ---- phase 1: pipelined M-tile loop ---------------------------------
    const int wave   = tid >> 5;
    const int lane   = tid & 31;
    const int lane15 = lane & 15;
    const int khalf  = (lane >> 4) << 1;   // lanes 0-15: K+0/1, lanes 16-31: K+2/3
    const int Kfloor = K & ~3;

    for (int mt = 0; mt < MTILES; ++mt) {
        if (wave < NCOMP) {
            // -------- compute wave: tile (mt, nt=wave) -------------------
            const int nt    = wave;
            const int fbase = mt << 4;
            const int tbase = nt << 4;
            const float* arow = &wbuf[mt & 1][lane15 * K + khalf];
            const float* brow = &xbuf[tbase + lane15 + khalf];

            v8f acc = {};
            #pragma unroll 4
            for (int k0 = 0; k0 < Kfloor; k0 += 4) {
                v2f a, b;
                a.x = arow[k0];
                a.y = arow[k0 + 1];
                b.x = brow[k0];
                b.y = brow[k0 + 1];
                acc = __builtin_amdgcn_wmma_f32_16x16x4_f32(
                    false, a, false, b, (short)0, acc, false, false);
            }
            if (Kfloor < K) {              // guarded tail (K % 4 != 0)
                const int kk = Kfloor + khalf;
                v2f a, b;
                a.x = (kk     < K) ? arow[Kfloor]     : 0.0f;
                a.y = (kk + 1 < K) ? arow[Kfloor + 1] : 0.0f;
                b.x = brow[Kfloor];        // xbuf zero-padded past K+NT
                b.y = brow[Kfloor + 1];
                acc = __builtin_amdgcn_wmma_f32_16x16x4_f32(
                    false, a, false, b, (short)0, acc, false, false);
            }

            // D layout: VGPR j -> M = j (+8 for lanes 16-31), N = lane15
            const int fout = fbase + ((lane >> 4) << 3);
            const int tcol = tbase + lane15;
            #pragma unroll
            for (int j = 0; j < 8; ++j)
                ytile[fout + j][tcol] = acc[j];
        } else if (mt + 1 < MTILES) {
            // -------- loader waves: prefetch W chunk mt+1 ----------------
            const int ltid = tid - NCOMP * 32;           // 0..95
            const int n4   = 4 * K;
            const float4* src = (const float4*)(w + (size_t)((mt + 1) << 4) * K);
            float4* dst = (float4*)&wbuf[(mt + 1) & 1][0];
            for (int i = ltid; i < n4; i += NTHR - NCOMP * 32) dst[i] = src[i];
        }
        __syncthreads();
    }

    // ---- phase 2: IMG_SELECT gather, coalesced stream-out ---------------
    // sel[s] = floor(s*71/223), exact in integer arithmetic (71,223 coprime).
    float* og = out + (size_t)bc * NFILT * TOUT;
    for (int idx = tid; idx < NFILT * TOUT; idx += NTHR) {
        const int f = idx / TOUT;
        const int s = idx - f * TOUT;
        const int t = (s * 71) / 223;
        og[idx] = ytile[f][t];
    }
}

extern "C" void kernel_launch(void* const* d_in, const int* in_sizes, int n_in,
                              void* d_out, int out_size, void* d_ws, size_t ws_size,
                              hipStream_t stream) {
    const float* x = (const float*)d_in[0];   // (32,12,2048) f32
    const float* w = (const float*)d_in[1];   // (112,1,K) f32
    float* out = (float*)d_out;               // (32,12,112,224) f32

    const int K   = in_sizes[1] / NFILT;      // filter length (561)
    const int pad = (K - 1) / 2;
    const int BC  = in_sizes[0] / LCONV;      // 384

    if (K > KMAX || K + NT > XBUF_N) return;  // static LDS bounds (never hit)

    if (K == 561) {
        cwt_wmma_kernel<561><<<dim3(BC), dim3(NTHR), 0, stream>>>(x, w, out, K, pad);
    } else {
        cwt_wmma_kernel<0><<<dim3(BC), dim3(NTHR), 0, stream>>>(x, w, out, K, pad);
    }
}